// DPCABlock_38929583571413
// MI455X (gfx1250) — compile-verified
//
#include <hip/hip_runtime.h>
#include <math.h>

// ---------------- problem constants ----------------
#define B_     4
#define C_     256
#define L_     4096          // 64*64
#define HEADS_ 8
#define DH_    32
#define BH_    32            // B_*HEADS_
#define TOPK_  512           // L_/HEADS_
#define INNER_ 256           // HEADS_*DH_
#define EPS_LN 1e-5f
#define EPS_L2 1e-12f

typedef __attribute__((ext_vector_type(2))) float        v2f;
typedef __attribute__((ext_vector_type(8))) float        v8f;
typedef __attribute__((ext_vector_type(4))) unsigned int v4u;
typedef __attribute__((ext_vector_type(8))) int          v8i;
typedef __attribute__((ext_vector_type(4))) int          v4i;

// fp32 WMMA: D(16x16) = A(16x4) * B(4x16) + C ; one issue per K=4 step.
__device__ __forceinline__ v8f wmma4(v2f a, v2f b, v8f c) {
  return __builtin_amdgcn_wmma_f32_16x16x4_f32(false, a, false, b, (short)0, c,
                                               false, false);
}

// ---------------------------------------------------------------------
// TDM: DMA a contiguous run of `n4` dwords from `gaddr` into LDS offset
// `lds_off` (bytes). D# built per CDNA5 ISA §8.3/8.4:
//   group0: count=1, lds_addr, global_addr[56:0], type=2
//   group1: data_size=4B, tensor_dim0=tile_dim0=n4, dim1=1, stride0=n4
// ---------------------------------------------------------------------
__device__ __forceinline__ void tdm_load_1d(const void* gaddr,
                                            unsigned lds_off, unsigned n4) {
  unsigned long long ga = (unsigned long long)gaddr;
  v4u g0;
  g0.x = 1u;                                           // count=1
  g0.y = lds_off;                                      // lds_addr (bytes)
  g0.z = (unsigned)(ga & 0xffffffffu);                 // global_addr lo
  g0.w = (unsigned)((ga >> 32) & 0x01ffffffu) | 0x80000000u; // hi | type=2
  v8i g1;
  g1[0] = 0x00020000;                    // data_size=2 (4 bytes)
  g1[1] = (int)((n4 & 0xffffu) << 16);   // tensor_dim0[15:0]
  g1[2] = (int)(((n4 >> 16) & 0xffffu) | (1u << 16)); // dim0 hi | tensor_dim1=1
  g1[3] = (int)((n4 & 0xffffu) << 16);   // tile_dim0 (tile_dim0[31:16] unused: n4<64K)
  g1[4] = 1;                             // tile_dim1=1, tile_dim2=0
  g1[5] = (int)n4;                       // tensor_dim0_stride lo32
  g1[6] = 0;
  g1[7] = 0;
  v4i z4 = {0, 0, 0, 0};
#if defined(__clang_major__) && (__clang_major__ >= 23)
  v8i z8 = {0, 0, 0, 0, 0, 0, 0, 0};
  __builtin_amdgcn_tensor_load_to_lds(g0, g1, z4, z4, z8, 0);
#else
  __builtin_amdgcn_tensor_load_to_lds(g0, g1, z4, z4, 0);
#endif
}

// =====================================================================
// 1) channel LayerNorm over C for (B,C,L) layout; one thread per pixel.
// =====================================================================
__global__ void k_chan_ln(const float* __restrict__ x,
                          const float* __restrict__ g,
                          const float* __restrict__ bb,
                          float* __restrict__ y) {
  int p = blockIdx.x * blockDim.x + threadIdx.x;   // 0..B_*L_
  int b = p >> 12, l = p & (L_ - 1);
  const float* xp = x + (size_t)b * C_ * L_ + l;
  float s = 0.f, s2 = 0.f;
  for (int c = 0; c < C_; ++c) {
    float v = xp[(size_t)c * L_];
    s += v; s2 += v * v;
  }
  float mu  = s * (1.0f / C_);
  float var = s2 * (1.0f / C_) - mu * mu;
  float r   = rsqrtf(var + EPS_LN);
  float* yp = y + (size_t)b * C_ * L_ + l;
  for (int c = 0; c < C_; ++c) {
    float v = xp[(size_t)c * L_];
    yp[(size_t)c * L_] = (v - mu) * r * g[c] + bb[c];
  }
}

// =====================================================================
// 2) fp32 WMMA tile: 16x16 C-tile of  W(M=o,K=c) x X(K=c,N=l), K=256.
//    Wide unroll -> compile-time k0 -> immediate-offset global loads.
// =====================================================================
__device__ __forceinline__ v8f gemm_tile_WX(const float* __restrict__ Wrow,
                                            const float* __restrict__ Xcol,
                                            int lane) {
  int half = lane >> 4, n = lane & 15;
  const float* wl = Wrow + n * C_;            // A row m = lane&15
  const float* xb = Xcol + n;                 // B col n
  v8f c = {};
#pragma unroll 16
  for (int j = 0; j < 64; ++j) {
    int k0 = 4 * j;
    v2f a = *(const v2f*)(wl + k0 + 2 * half);
    v2f b;
    b.x = xb[(size_t)(k0 + half) * L_];
    b.y = xb[(size_t)(k0 + 2 + half) * L_];
    c = wmma4(a, b, c);
  }
  return c;
}

// q projection: q_raw scattered to head-major q_f[bh][l][d]
__global__ void k_proj_q(const float* __restrict__ Wq,
                         const float* __restrict__ qs_n,
                         float* __restrict__ q_f) {
  int lane = threadIdx.x;
  int lt = blockIdx.x * 16, ot = blockIdx.y * 16, b = blockIdx.z;
  v8f c = gemm_tile_WX(Wq + (size_t)ot * C_,
                       qs_n + (size_t)b * C_ * L_ + lt, lane);
  int half = lane >> 4, n = lane & 15, token = lt + n;
#pragma unroll
  for (int r = 0; r < 8; ++r) {
    int o = ot + r + half * 8;
    q_f[(((size_t)(b * HEADS_ + (o >> 5)) * L_ + token) * DH_) + (o & 31)] = c[r];
  }
}

// kv projection: rows [0,256) -> k_f, [256,512) -> v_f (head-major)
__global__ void k_proj_kv(const float* __restrict__ Wkv,
                          const float* __restrict__ ctx_n,
                          float* __restrict__ k_f,
                          float* __restrict__ v_f) {
  int lane = threadIdx.x;
  int lt = blockIdx.x * 16, ot = blockIdx.y * 16, b = blockIdx.z;
  v8f c = gemm_tile_WX(Wkv + (size_t)ot * C_,
                       ctx_n + (size_t)b * C_ * L_ + lt, lane);
  int half = lane >> 4, n = lane & 15, token = lt + n;
  float* dst = (ot < 256) ? k_f : v_f;
  int obase = (ot < 256) ? ot : (ot - 256);
#pragma unroll
  for (int r = 0; r < 8; ++r) {
    int o = obase + r + half * 8;
    dst[(((size_t)(b * HEADS_ + (o >> 5)) * L_ + token) * DH_) + (o & 31)] = c[r];
  }
}

// =====================================================================
// 3) l2norm over last dim (32) for (BH*L, 32) vectors, in place.
// =====================================================================
__global__ void k_l2norm(float* __restrict__ t) {
  int i = blockIdx.x * blockDim.x + threadIdx.x;   // BH_*L_
  float* p = t + (size_t)i * DH_;
  float s = 0.f;
#pragma unroll
  for (int j = 0; j < DH_; ++j) s += p[j] * p[j];
  float inv = 1.0f / fmaxf(sqrtf(s), EPS_L2);
#pragma unroll
  for (int j = 0; j < DH_; ++j) p[j] *= inv;
}

// =====================================================================
// 4) gather Q_small = q_f[bh, rand_token_ind]
// =====================================================================
__global__ void k_gather_q(const float* __restrict__ q_f,
                           const int* __restrict__ rti,
                           float* __restrict__ q_small) {
  int i = blockIdx.x * blockDim.x + threadIdx.x;   // BH_*TOPK_
  int bh = i >> 9;
  int tok = rti[i];
  const float* src = q_f + ((size_t)bh * L_ + tok) * DH_;
  float* dst = q_small + (size_t)i * DH_;
#pragma unroll
  for (int d = 0; d < DH_; ++d) dst[d] = src[d];
}

// =====================================================================
// 5) min over queries of L1(K,Q): Q_small DMA'd LDS-resident via TDM.
// =====================================================================
__global__ void k_min_dist(const float* __restrict__ k_f,
                           const float* __restrict__ q_small,
                           float* __restrict__ min_dist) {
  extern __shared__ float qs[];                    // 512*32 floats = 64KB @ LDS 0
  int bh = blockIdx.x, tid = threadIdx.x;
  if (tid < 32) {                                  // wave 0 drives the TDM
    tdm_load_1d(q_small + (size_t)bh * TOPK_ * DH_, 0u,
                (unsigned)(TOPK_ * DH_));
    __builtin_amdgcn_s_wait_tensorcnt(0);
  }
  __syncthreads();
  for (int t = 0; t < 16; ++t) {
    int tok = t * 256 + tid;
    const float* kp = k_f + ((size_t)bh * L_ + tok) * DH_;
    float kv[DH_];
#pragma unroll
    for (int j = 0; j < DH_; ++j) kv[j] = kp[j];
    float best = 3.0e38f;
    for (int r = 0; r < TOPK_; ++r) {
      const float* qp = qs + r * DH_;
      float s = 0.f;
#pragma unroll
      for (int j = 0; j < DH_; ++j) s += fabsf(kv[j] - qp[j]);
      best = fminf(best, s);
    }
    min_dist[(size_t)bh * L_ + tok] = best;
  }
}

// =====================================================================
// 6) deterministic top-512 by rank counting (value, index) order.
// =====================================================================
__global__ void k_select(const float* __restrict__ min_dist,
                         int* __restrict__ sel_idx) {
  extern __shared__ float sd[];                    // 4096 floats
  int bh = blockIdx.x, tid = threadIdx.x;
  const float* db = min_dist + (size_t)bh * L_;
  for (int u = tid; u < L_; u += blockDim.x) sd[u] = db[u];
  __syncthreads();
  for (int t = 0; t < 16; ++t) {
    int i = t * 256 + tid;
    float di = sd[i];
    int rank = 0;
    for (int j = 0; j < L_; ++j) {
      float dj = sd[j];
      rank += (dj < di) || (dj == di && j < i);
    }
    if (rank < TOPK_) sel_idx[bh * TOPK_ + rank] = i;
  }
}

// =====================================================================
// 7) gather k_sel / v_sel (contiguous [bh][k][d] for WMMA B loads)
// =====================================================================
__global__ void k_gather_kv(const float* __restrict__ k_f,
                            const float* __restrict__ v_f,
                            const int* __restrict__ sel_idx,
                            float* __restrict__ k_sel,
                            float* __restrict__ v_sel) {
  int i = blockIdx.x * blockDim.x + threadIdx.x;   // BH_*TOPK_
  int bh = i >> 9;
  int tok = sel_idx[i];
  const float* ks = k_f + ((size_t)bh * L_ + tok) * DH_;
  const float* vs = v_f + ((size_t)bh * L_ + tok) * DH_;
  float* kd = k_sel + (size_t)i * DH_;
  float* vd = v_sel + (size_t)i * DH_;
#pragma unroll
  for (int d = 0; d < DH_; ++d) { kd[d] = ks[d]; vd[d] = vs[d]; }
}

// =====================================================================
// 8) flash attention: per-wave 16-query tile, K=512 keys in 32 chunks.
//    QK^T and PV both on v_wmma_f32_16x16x4_f32; online softmax with
//    shfl_xor row reductions in wave32 halves; next chunk prefetched.
// =====================================================================
#define ATT_WAVES 8
__global__ void k_attn(const float* __restrict__ q_f,
                       const float* __restrict__ k_sel,
                       const float* __restrict__ v_sel,
                       float* __restrict__ att) {
  __shared__ float kch[16 * DH_];
  __shared__ float vch[16 * DH_];
  __shared__ float pl[ATT_WAVES][16][16];
  int tid = threadIdx.x;
  int wave = tid >> 5, lane = tid & 31;
  int half = lane >> 4, n = lane & 15;
  int bh = blockIdx.y, b = bh >> 3, h = bh & 7;
  int tile0 = (blockIdx.x * ATT_WAVES + wave) * 16;

  // preload Q A-operands: 8 K-steps of 4
  v2f aq[8];
  const float* qrow = q_f + ((size_t)bh * L_ + tile0 + n) * DH_;
#pragma unroll
  for (int j = 0; j < 8; ++j) aq[j] = *(const v2f*)(qrow + 4 * j + 2 * half);

  v8f acc0 = {}, acc1 = {};
  float rmax[8], rsum[8];
#pragma unroll
  for (int r = 0; r < 8; ++r) { rmax[r] = -3.0e38f; rsum[r] = 0.f; }

  const float* kb = k_sel + (size_t)bh * TOPK_ * DH_;
  const float* vb = v_sel + (size_t)bh * TOPK_ * DH_;

  for (int ch = 0; ch < 32; ++ch) {
    __syncthreads();
    for (int u = tid; u < 16 * DH_; u += 256) {
      kch[u] = kb[ch * 16 * DH_ + u];
      vch[u] = vb[ch * 16 * DH_ + u];
    }
    if (ch < 31) {                     // global_prefetch_b8 next chunk
      __builtin_prefetch(kb + (ch + 1) * 16 * DH_ + tid * 2, 0, 3);
      __builtin_prefetch(vb + (ch + 1) * 16 * DH_ + tid * 2, 0, 3);
    }
    __syncthreads();

    // scores tile: B[k=dim][n=key] = kch[key*32+dim]
    v8f s = {};
#pragma unroll
    for (int j = 0; j < 8; ++j) {
      v2f bv;
      bv.x = kch[n * DH_ + 4 * j + half];
      bv.y = kch[n * DH_ + 4 * j + 2 + half];
      s = wmma4(aq[j], bv, s);
    }

    // online softmax per row (register r -> row r+8*half)
    float p[8];
#pragma unroll
    for (int r = 0; r < 8; ++r) {
      float cm = s[r];
      for (int m = 1; m < 16; m <<= 1) cm = fmaxf(cm, __shfl_xor(cm, m, 32));
      float nmax  = fmaxf(rmax[r], cm);
      float pe    = __expf(s[r] - nmax);
      float cs    = pe;
      for (int m = 1; m < 16; m <<= 1) cs += __shfl_xor(cs, m, 32);
      float scale = __expf(rmax[r] - nmax);
      rsum[r] = rsum[r] * scale + cs;
      rmax[r] = nmax;
      acc0[r] *= scale;
      acc1[r] *= scale;
      p[r] = pe;
    }

    // C-layout -> A-layout through private wave slab (in-wave LDS in-order)
#pragma unroll
    for (int r = 0; r < 8; ++r) pl[wave][r + half * 8][n] = p[r];

    // out += P(16x16) * V(16x32): 4 K-steps x 2 N-tiles
#pragma unroll
    for (int j = 0; j < 4; ++j) {
      v2f ap;
      ap.x = pl[wave][n][4 * j + 2 * half];
      ap.y = pl[wave][n][4 * j + 2 * half + 1];
      int kr0 = 4 * j + half, kr1 = 4 * j + 2 + half;
      v2f bv0, bv1;
      bv0.x = vch[kr0 * DH_ + n];       bv0.y = vch[kr1 * DH_ + n];
      bv1.x = vch[kr0 * DH_ + 16 + n];  bv1.y = vch[kr1 * DH_ + 16 + n];
      acc0 = wmma4(ap, bv0, acc0);
      acc1 = wmma4(ap, bv1, acc1);
    }
  }

  // finalize: divide by rowsum, scatter to att[b][l][h*32+d]
#pragma unroll
  for (int r = 0; r < 8; ++r) {
    float inv = 1.0f / rsum[r];
    int token = tile0 + r + half * 8;
    float* op = att + ((size_t)b * L_ + token) * INNER_ + h * DH_;
    op[n]      = acc0[r] * inv;
    op[16 + n] = acc1[r] * inv;
  }
}

// =====================================================================
// 9) Wo projection: proj[b][l][o] = sum_i Wo[o][i]*att[b][l][i]
// =====================================================================
__global__ void k_proj_o(const float* __restrict__ Wo,
                         const float* __restrict__ att,
                         float* __restrict__ proj) {
  int lane = threadIdx.x;
  int half = lane >> 4, n = lane & 15;
  int lt = blockIdx.x * 16, ot = blockIdx.y * 16, b = blockIdx.z;
  const float* arow = att + ((size_t)b * L_ + lt + n) * INNER_;
  const float* wcol = Wo + (size_t)(ot + n) * INNER_;
  v8f c = {};
#pragma unroll 16
  for (int j = 0; j < 64; ++j) {
    int k0 = 4 * j;
    v2f a = *(const v2f*)(arow + k0 + 2 * half);
    v2f bb;
    bb.x = wcol[k0 + half];
    bb.y = wcol[k0 + 2 + half];
    c = wmma4(a, bb, c);
  }
#pragma unroll
  for (int r = 0; r < 8; ++r) {
    int token = lt + r + half * 8;
    proj[((size_t)b * L_ + token) * INNER_ + ot + n] = c[r];
  }
}

// =====================================================================
// 10) final LN over o + gamma*out + residual, write (B,C,L) layout.
// =====================================================================
__global__ void k_final(const float* __restrict__ proj,
                        const float* __restrict__ g,
                        const float* __restrict__ bb,
                        const float* __restrict__ gamma,
                        const float* __restrict__ qsrc,
                        float* __restrict__ out) {
  int p = blockIdx.x * blockDim.x + threadIdx.x;   // B_*L_
  int b = p >> 12, l = p & (L_ - 1);
  const float* pp = proj + (size_t)p * INNER_;
  float s = 0.f, s2 = 0.f;
  for (int o = 0; o < INNER_; ++o) { float v = pp[o]; s += v; s2 += v * v; }
  float mu  = s * (1.0f / INNER_);
  float var = s2 * (1.0f / INNER_) - mu * mu;
  float r   = rsqrtf(var + EPS_LN);
  float gm  = gamma[0];
  for (int o = 0; o < INNER_; ++o) {
    float v = (pp[o] - mu) * r * g[o] + bb[o];
    size_t oi = ((size_t)b * C_ + o) * L_ + l;
    out[oi] = gm * v + qsrc[oi];
  }
}

// =====================================================================
extern "C" void kernel_launch(void* const* d_in, const int* in_sizes, int n_in,
                              void* d_out, int out_size, void* d_ws, size_t ws_size,
                              hipStream_t stream) {
  const float* qsrc   = (const float*)d_in[0];
  const float* ctx    = (const float*)d_in[1];
  const int*   rti    = (const int*)  d_in[2];
  const float* g_ctx  = (const float*)d_in[3];
  const float* b_ctx  = (const float*)d_in[4];
  const float* g_qs   = (const float*)d_in[5];
  const float* b_qs   = (const float*)d_in[6];
  const float* g_out  = (const float*)d_in[7];
  const float* b_out  = (const float*)d_in[8];
  const float* Wq     = (const float*)d_in[9];
  const float* Wkv    = (const float*)d_in[10];
  const float* Wo     = (const float*)d_in[11];
  const float* gamma  = (const float*)d_in[12];
  float* out = (float*)d_out;

  const size_t PLANE = (size_t)B_ * C_ * L_;       // 4,194,304 floats
  float* ws      = (float*)d_ws;
  float* ctx_n   = ws;
  float* qs_n    = ws + PLANE;
  float* q_f     = ws + 2 * PLANE;
  float* k_f     = ws + 3 * PLANE;
  float* v_f     = ws + 4 * PLANE;
  float* extra   = ws + 5 * PLANE;
  float* q_small = extra;                           // 524288
  float* k_sel   = extra + 524288;                  // 524288
  float* v_sel   = extra + 1048576;                 // 524288
  float* min_d   = extra + 1572864;                 // 131072
  int*   sel_idx = (int*)(extra + 1703936);         // 16384 ints
  float* att     = ctx_n;                           // reuse (dead after proj_kv)
  float* proj    = qs_n;                            // reuse (dead after proj_q)

  const int PIX_BLOCKS = (B_ * L_) / 256;           // 64

  // 1-2) channel LayerNorms
  k_chan_ln<<<PIX_BLOCKS, 256, 0, stream>>>(ctx,  g_ctx, b_ctx, ctx_n);
  k_chan_ln<<<PIX_BLOCKS, 256, 0, stream>>>(qsrc, g_qs,  b_qs,  qs_n);

  // 3-4) WMMA projections (one wave per 16x16 tile)
  k_proj_kv<<<dim3(L_ / 16, 32, B_), 32, 0, stream>>>(Wkv, ctx_n, k_f, v_f);
  k_proj_q <<<dim3(L_ / 16, 16, B_), 32, 0, stream>>>(Wq,  qs_n,  q_f);

  // 5) l2norm q, k
  k_l2norm<<<(BH_ * L_) / 256, 256, 0, stream>>>(q_f);
  k_l2norm<<<(BH_ * L_) / 256, 256, 0, stream>>>(k_f);

  // 6) gather Q_small
  k_gather_q<<<(BH_ * TOPK_) / 256, 256, 0, stream>>>(q_f, rti, q_small);

  // 7) L1 min-distance scan (Q_small TDM-DMA'd into 64KB dyn LDS)
  k_min_dist<<<BH_, 256, TOPK_ * DH_ * sizeof(float), stream>>>(k_f, q_small, min_d);

  // 8) deterministic top-512 rank select
  k_select<<<BH_, 256, L_ * sizeof(float), stream>>>(min_d, sel_idx);

  // 9) gather selected K/V
  k_gather_kv<<<(BH_ * TOPK_) / 256, 256, 0, stream>>>(k_f, v_f, sel_idx, k_sel, v_sel);

  // 10) flash attention (8 waves/block, 16 query rows per wave)
  k_attn<<<dim3(L_ / (16 * ATT_WAVES), BH_), 256, 0, stream>>>(q_f, k_sel, v_sel, att);

  // 11) output projection
  k_proj_o<<<dim3(L_ / 16, 16, B_), 32, 0, stream>>>(Wo, att, proj);

  // 12) final LN + gamma*out + residual
  k_final<<<PIX_BLOCKS, 256, 0, stream>>>(proj, g_out, b_out, gamma, qsrc, out);
}